// GE2ELoss_80711025427125
// MI455X (gfx1250) — compile-verified
//
#include <hip/hip_runtime.h>
#include <hip/hip_bf16.h>
#include <math.h>

// GE2E loss for dvectors[S=1024, U=32, D=256] f32, outputs (loss, pos_mean, neg_mean).
// Core GEMM all_cos = dn @ cn^T done with V_WMMA_F32_16X16X4_F32 on raw (unnormalized)
// data; cosines recovered by per-row/per-col inverse-norm scaling. LSE is computed
// online per-lane so the 32768x1024 matrix is never materialized.
//
// v2: A panel (16x256) held entirely in VGPRs (64 v2f frags); B fragments
// double-buffered in batches of 8 so global loads overlap the WMMA pipe
// instead of the previous load -> s_wait_loadcnt 0 -> wmma serialization.

#define S_SPK 1024
#define U_UTT 32
#define D_DIM 256
#define M_ROWS (S_SPK * U_UTT)   // 32768

typedef __attribute__((ext_vector_type(2))) float v2f;
typedef __attribute__((ext_vector_type(8))) float v8f;

// ---------------- workspace layout (floats) ----------------
// [0..3]   accumulators: 0=pos_sum 1=lse_sum 2=allcos_sum 3=own_sum
// [16 ..]                     ssum[s][d]   speaker sums (B matrix, row-major 1024x256)
// [WS_COLINV ..]              colinv[s]    1/max(|ssum_s|, U*1e-8)
// [WS_ROWINV ..]              rowinv[m]    1/max(|x_m|, 1e-8)
#define WS_ACC    0
#define WS_SSUM   16
#define WS_COLINV (16 + S_SPK * D_DIM)
#define WS_ROWINV (WS_COLINV + S_SPK)

__device__ __forceinline__ float wave_sum32(float v) {
    #pragma unroll
    for (int o = 1; o < 32; o <<= 1) v += __shfl_xor(v, o, 32);
    return v;
}

__global__ void ge2e_init(float* acc) {
    if (threadIdx.x < 4) acc[threadIdx.x] = 0.0f;
}

// One block per speaker, 256 threads (8 waves).
// Produces: ssum row, colinv, rowinv per utterance, and atomically accumulates
// sum of leave-one-out positive cosines.
__global__ void __launch_bounds__(256) ge2e_speaker_stats(
        const float* __restrict__ dv, float* __restrict__ ws) {
    __shared__ float sv[U_UTT * D_DIM];  // 32 KB: this speaker's dvectors
    __shared__ float ssumS[D_DIM];       // column sums over utterances
    __shared__ float part[8];

    const int s    = blockIdx.x;
    const int tid  = threadIdx.x;
    const int lane = tid & 31;
    const int wv   = tid >> 5;
    const float* base = dv + (size_t)s * (U_UTT * D_DIM);

    // load speaker tile
    for (int i = tid; i < U_UTT * D_DIM; i += 256) sv[i] = base[i];
    __syncthreads();

    // per-dim column sum over the 32 utterances
    float cs = 0.0f;
    #pragma unroll
    for (int u = 0; u < U_UTT; ++u) cs += sv[u * D_DIM + tid];
    ssumS[tid] = cs;
    ws[WS_SSUM + (size_t)s * D_DIM + tid] = cs;

    // |ssum|^2 block reduction -> colinv
    float q = wave_sum32(cs * cs);
    if (lane == 0) part[wv] = q;
    __syncthreads();
    if (tid == 0) {
        float qs = 0.0f;
        #pragma unroll
        for (int i = 0; i < 8; ++i) qs += part[i];
        // cos(x, cent) = (x . ssum) / (max(|x|,eps8) * max(|ssum|, U*eps8))
        ws[WS_COLINV + s] = 1.0f / fmaxf(sqrtf(qs), (float)U_UTT * 1e-8f);
    }

    // per-utterance: leave-one-out positive cosine + row inverse norms.
    // wave wv handles utterances wv, wv+8, wv+16, wv+24.
    float posacc = 0.0f;
    for (int u = wv; u < U_UTT; u += 8) {
        float dd = 0.0f, pp = 0.0f, dp = 0.0f;
        #pragma unroll
        for (int i = 0; i < D_DIM / 32; ++i) {
            int d = lane + i * 32;
            float x = sv[u * D_DIM + d];
            float p = ssumS[d] - x;       // 31 * pos_centroid
            dd += x * x;
            pp += p * p;
            dp += x * p;
        }
        dd = wave_sum32(dd);
        pp = wave_sum32(pp);
        dp = wave_sum32(dp);
        if (lane == 0) {
            const float invU1 = 1.0f / (float)(U_UTT - 1);
            float nx = fmaxf(sqrtf(dd), 1e-6f);            // EPS_POS
            float np = fmaxf(sqrtf(pp) * invU1, 1e-6f);    // EPS_POS on true centroid
            posacc += (dp * invU1) / (nx * np);
            ws[WS_ROWINV + (size_t)s * U_UTT + u] = 1.0f / fmaxf(sqrtf(dd), 1e-8f);
        }
    }
    if (lane == 0) atomicAdd(&ws[WS_ACC + 0], posacc);
}

// GEMM + online logsumexp. 128 threads = 4 waves per block; each wave owns a
// 16-row block of A (raw dvectors, preloaded to VGPRs) and sweeps all 64
// column tiles of B (raw speaker sums), K=256 via 64x V_WMMA_F32_16X16X4_F32
// per tile with double-buffered B fragment loads.
__global__ void __launch_bounds__(128) ge2e_gemm_lse(
        const float* __restrict__ dv, const float* __restrict__ wp,
        const float* __restrict__ bp, float* __restrict__ ws) {
    const int lane = threadIdx.x & 31;
    const int wv   = threadIdx.x >> 5;
    const int gw   = blockIdx.x * 4 + wv;      // 2048 waves total
    const int m0   = gw * 16;                  // 16-row block (inside one speaker)
    const int s_own = m0 >> 5;                 // own speaker for all 16 rows
    const int lh   = lane & 15;
    const int half = lane >> 4;                // 0: rows m0..m0+7, 1: rows m0+8..m0+15
    const int koff = half * 2;                 // K sub-offset per WMMA A/B layout

    const float* ssum   = ws + WS_SSUM;
    const float* colinv = ws + WS_COLINV;
    const float* rowinv = ws + WS_ROWINV;

    const float w0 = wp[0];
    const float b0 = bp[0];

    // inverse norms of the 8 rows this half-wave accumulates
    float rinv[8];
    #pragma unroll
    for (int j = 0; j < 8; ++j) rinv[j] = rowinv[m0 + half * 8 + j];

    // ---- preload the entire 16x256 A panel into registers (64 v2f) ----
    const float* arow = dv + (size_t)(m0 + lh) * D_DIM + koff;
    v2f afr[64];
    #pragma unroll
    for (int i = 0; i < 64; ++i) afr[i] = *(const v2f*)(arow + 4 * i);

    const float NEG_INF = -__builtin_inff();
    float lm[8], ls[8];
    #pragma unroll
    for (int j = 0; j < 8; ++j) { lm[j] = NEG_INF; ls[j] = 0.0f; }
    float allacc = 0.0f, ownacc = 0.0f;

    for (int n0 = 0; n0 < S_SPK; n0 += 16) {
        const float* brow = ssum + (size_t)(n0 + lh) * D_DIM + koff;
        const float cinv  = colinv[n0 + lh];
        const bool  isown = (n0 + lh) == s_own;

        v8f c = {0.0f, 0.0f, 0.0f, 0.0f, 0.0f, 0.0f, 0.0f, 0.0f};

        // double-buffered B fragments, 8 at a time
        v2f bb0[8], bb1[8];
        #pragma unroll
        for (int i = 0; i < 8; ++i) bb0[i] = *(const v2f*)(brow + 4 * i);

        #pragma unroll
        for (int kk = 0; kk < 64; kk += 8) {
            const bool even = ((kk >> 3) & 1) == 0;
            if (kk + 8 < 64) {
                if (even) {
                    #pragma unroll
                    for (int i = 0; i < 8; ++i)
                        bb1[i] = *(const v2f*)(brow + 4 * (kk + 8 + i));
                } else {
                    #pragma unroll
                    for (int i = 0; i < 8; ++i)
                        bb0[i] = *(const v2f*)(brow + 4 * (kk + 8 + i));
                }
            }
            #pragma unroll
            for (int i = 0; i < 8; ++i) {
                v2f b = even ? bb0[i] : bb1[i];
                c = __builtin_amdgcn_wmma_f32_16x16x4_f32(
                        false, afr[kk + i], false, b, (short)0, c, false, false);
            }
        }

        #pragma unroll
        for (int j = 0; j < 8; ++j) {
            float cosv = c[j] * rinv[j] * cinv;   // true cosine
            allacc += cosv;
            if (isown) {
                ownacc += cosv;                   // excluded from LSE
            } else {
                float t  = w0 * cosv + b0;
                float nm = fmaxf(lm[j], t);
                ls[j] = ls[j] * expf(lm[j] - nm) + expf(t - nm);
                lm[j] = nm;
            }
        }
    }

    // merge the 16 per-lane LSE states of each half-wave, sum lse over 8 rows
    float lsesum = 0.0f;
    #pragma unroll
    for (int j = 0; j < 8; ++j) {
        float m = lm[j], s = ls[j];
        #pragma unroll
        for (int o = 1; o < 16; o <<= 1) {
            float om = __shfl_xor(m, o, 32);
            float os = __shfl_xor(s, o, 32);
            float nm = fmaxf(m, om);
            s = s * expf(m - nm) + os * expf(om - nm);
            m = nm;
        }
        lsesum += m + logf(s);
    }
    lsesum += __shfl_xor(lsesum, 16, 32);   // add the other half's 8 rows

    allacc = wave_sum32(allacc);
    ownacc = wave_sum32(ownacc);
    if (lane == 0) {
        atomicAdd(&ws[WS_ACC + 1], lsesum);
        atomicAdd(&ws[WS_ACC + 2], allacc);
        atomicAdd(&ws[WS_ACC + 3], ownacc);
    }
}

__global__ void ge2e_finalize(const float* __restrict__ ws,
                              const float* __restrict__ wp,
                              const float* __restrict__ bp,
                              float* __restrict__ out) {
    const float NSU = (float)(S_SPK * U_UTT);
    float pos_mean = ws[WS_ACC + 0] / NSU;
    float lse_mean = ws[WS_ACC + 1] / NSU;
    float neg_mean = (ws[WS_ACC + 2] - ws[WS_ACC + 3]) /
                     (NSU * (float)(S_SPK - 1));
    float loss = -(wp[0] * pos_mean + bp[0]) + lse_mean;
    out[0] = loss;
    out[1] = pos_mean;
    out[2] = neg_mean;
}

extern "C" void kernel_launch(void* const* d_in, const int* in_sizes, int n_in,
                              void* d_out, int out_size, void* d_ws, size_t ws_size,
                              hipStream_t stream) {
    (void)in_sizes; (void)n_in; (void)out_size; (void)ws_size;
    const float* dv = (const float*)d_in[0];
    const float* wp = (const float*)d_in[1];
    const float* bp = (const float*)d_in[2];
    float* out = (float*)d_out;
    float* ws  = (float*)d_ws;

    ge2e_init<<<1, 32, 0, stream>>>(ws);
    ge2e_speaker_stats<<<S_SPK, 256, 0, stream>>>(dv, ws);
    ge2e_gemm_lse<<<M_ROWS / 16 / 4, 128, 0, stream>>>(dv, wp, bp, ws);
    ge2e_finalize<<<1, 1, 0, stream>>>(ws, wp, bp, out);
}